// CRF_16569983828000
// MI455X (gfx1250) — compile-verified
//
#include <hip/hip_runtime.h>
#include <hip/hip_bf16.h>

typedef float v2f __attribute__((ext_vector_type(2)));
typedef float v8f __attribute__((ext_vector_type(8)));
typedef float f4  __attribute__((ext_vector_type(4)));

constexpr int BB = 64;      // batch
constexpr int TT = 512;     // time steps
constexpr int FF = 1024;    // features (K)
constexpr int UU = 64;      // tags (N)
constexpr int MM = BB * TT; // 32768 GEMM rows

// ---------------------------------------------------------------------------
// Kernel 1: potentials = x @ kernel + bias (+ boundary energies fused)
// fp32 WMMA 16x16x4. Block = 256 threads (8 waves); wave -> 16 M-rows x 64 N.
// K staged in 64-wide chunks through LDS (A padded stride 66, B transposed).
// ---------------------------------------------------------------------------
#define KCH 64
#define LDA 66
#define LDB 66

__global__ __launch_bounds__(256) void crf_potentials_gemm(
    const float* __restrict__ x, const float* __restrict__ w,
    const float* __restrict__ bias, const float* __restrict__ lb,
    const float* __restrict__ rb, float* __restrict__ pot) {
  __shared__ __attribute__((aligned(16))) float lds_a[128 * LDA]; // 33.8 KB
  __shared__ __attribute__((aligned(16))) float lds_b[64 * LDB];  // 16.9 KB

  const int tid = threadIdx.x;
  const int lane = tid & 31;
  const int wave = tid >> 5;
  const int blockRow = blockIdx.x * 128;
  const int rowBase = blockRow + wave * 16;

  const int mrow = lane & 15;            // M (and N) index within tile
  const int kbase = (lane >> 4) << 1;    // lanes 16-31 hold K+2,K+3

  v8f acc[4] = {};

  for (int kk = 0; kk < FF; kk += KCH) {
    __syncthreads();
    // --- stage A tile: 128 rows x 64 K (coalesced b128 loads) ---
#pragma unroll
    for (int i = 0; i < 8; ++i) {
      int f = tid + i * 256;   // float4 index in [0,2048)
      int row = f >> 4;
      int q = f & 15;
      const float* src = x + (size_t)(blockRow + row) * FF + kk + q * 4;
      if (kk + KCH < FF) __builtin_prefetch(src + KCH, 0, 0);
      f4 av = *(const f4*)src;
      float* dst = &lds_a[row * LDA + q * 4];
      dst[0] = av.x; dst[1] = av.y; dst[2] = av.z; dst[3] = av.w;
    }
    // --- stage B tile transposed: lds_b[n*LDB + k] = w[kk+k][n] ---
#pragma unroll
    for (int i = 0; i < 4; ++i) {
      int f = tid + i * 256;   // float4 index in [0,1024)
      int row = f >> 4;        // k within chunk
      int q = f & 15;          // n/4
      f4 bv = *(const f4*)(w + (size_t)(kk + row) * UU + q * 4);
      lds_b[(q * 4 + 0) * LDB + row] = bv.x;
      lds_b[(q * 4 + 1) * LDB + row] = bv.y;
      lds_b[(q * 4 + 2) * LDB + row] = bv.z;
      lds_b[(q * 4 + 3) * LDB + row] = bv.w;
    }
    __syncthreads();
    // --- 16 WMMA K-steps of 4 over this chunk ---
#pragma unroll
    for (int k4 = 0; k4 < KCH; k4 += 4) {
      v2f afrag = *(const v2f*)&lds_a[(wave * 16 + mrow) * LDA + k4 + kbase];
#pragma unroll
      for (int nt = 0; nt < 4; ++nt) {
        int n = nt * 16 + mrow;
        v2f bfrag = *(const v2f*)&lds_b[n * LDB + k4 + kbase];
        acc[nt] = __builtin_amdgcn_wmma_f32_16x16x4_f32(
            false, afrag, false, bfrag, (short)0, acc[nt], false, false);
      }
    }
  }

  // --- epilogue: + bias, + boundary energies, store potentials ---
  const int rsel = (lane >> 4) * 8; // C/D layout: VGPR r -> M=r (lo) / r+8 (hi)
#pragma unroll
  for (int nt = 0; nt < 4; ++nt) {
    int col = nt * 16 + mrow;
    float bia = bias[col];
    float lbv = lb[col];
    float rbv = rb[col];
#pragma unroll
    for (int r = 0; r < 8; ++r) {
      int row = rowBase + r + rsel;
      float v = acc[nt][r] + bia;
      int t = row & (TT - 1);
      if (t == 0) v += lbv;
      if (t == TT - 1) v += rbv;
      pot[(size_t)row * UU + col] = v;
    }
  }
}

// ---------------------------------------------------------------------------
// Kernel 2: Viterbi forward + backtrack. One block per batch element.
// 256 threads = 64 tags x 4 prev-groups (16 prev-tags each) + LDS reduce.
// Backpointers kept in LDS (32 KB); potentials double-buffered through LDS.
// ---------------------------------------------------------------------------
__global__ __launch_bounds__(256) void crf_viterbi(
    const float* __restrict__ pot, const float* __restrict__ chain,
    int* __restrict__ path) {
  __shared__ float alpha[UU];
  __shared__ float pv[4][UU];
  __shared__ int pidx[4][UU];
  __shared__ unsigned char bp[TT * UU];                          // 32 KB
  __shared__ __attribute__((aligned(16))) float potbuf[2][32 * UU]; // 16 KB

  const int tid = threadIdx.x;
  const int c = tid & 63;   // current tag
  const int pg = tid >> 6;  // prev-tag group 0..3
  const int b = blockIdx.x;
  const float* potb = pot + (size_t)b * TT * UU;

  // chain column slice: chain[p][c] for p in [pg*16, pg*16+16)
  float ck[16];
#pragma unroll
  for (int j = 0; j < 16; ++j) ck[j] = chain[(pg * 16 + j) * UU + c];

  // stage chunk 0 (t = 0..31)
#pragma unroll
  for (int i = 0; i < 2; ++i) {
    int f = tid + i * 256; // float4 index in [0,512)
    *(f4*)&potbuf[0][f * 4] = *(const f4*)(potb + f * 4);
  }
  __syncthreads();
  if (pg == 0) alpha[c] = potbuf[0][c]; // t=0 row (left boundary already fused)
  __syncthreads();

  int t = 1;
  for (int chunk = 0; chunk < 16; ++chunk) {
    const int buf = chunk & 1;
    // prefetch next chunk into registers (overlaps the serial chain of up to 32 steps)
    f4 nv0, nv1;
    if (chunk < 15) {
      nv0 = *(const f4*)(potb + (chunk + 1) * 32 * UU + (tid + 0) * 4);
      nv1 = *(const f4*)(potb + (chunk + 1) * 32 * UU + (tid + 256) * 4);
    }
    const int tEnd = (chunk + 1) * 32;
    for (; t < tEnd; ++t) {
      const int tl = t - chunk * 32;
      float best = -3.402823466e38f;
      int bestp = 0;
#pragma unroll
      for (int j = 0; j < 16; ++j) {
        float s = alpha[pg * 16 + j] + ck[j];
        if (s > best) { best = s; bestp = pg * 16 + j; }  // first-max tiebreak
      }
      pv[pg][c] = best;
      pidx[pg][c] = bestp;
      __syncthreads();
      if (pg == 0) {
        float bb = pv[0][c];
        int bi = pidx[0][c];
#pragma unroll
        for (int g = 1; g < 4; ++g) {
          float s = pv[g][c];
          if (s > bb) { bb = s; bi = pidx[g][c]; } // ascending order => first max
        }
        bp[t * UU + c] = (unsigned char)bi;
        alpha[c] = bb + potbuf[buf][tl * UU + c];
      }
      __syncthreads();
    }
    if (chunk < 15) {
      *(f4*)&potbuf[1 - buf][(tid + 0) * 4] = nv0;
      *(f4*)&potbuf[1 - buf][(tid + 256) * 4] = nv1;
      __syncthreads();
    }
  }

  // final argmax + backtrack entirely from LDS
  if (tid == 0) {
    float bb = alpha[0];
    int tag = 0;
    for (int i = 1; i < UU; ++i) {
      float s = alpha[i];
      if (s > bb) { bb = s; tag = i; }
    }
    int* outb = path + (size_t)b * TT;
    outb[TT - 1] = tag;
    for (int tt = TT - 1; tt >= 1; --tt) {
      tag = bp[tt * UU + tag];
      outb[tt - 1] = tag;
    }
  }
}

// ---------------------------------------------------------------------------
extern "C" void kernel_launch(void* const* d_in, const int* in_sizes, int n_in,
                              void* d_out, int out_size, void* d_ws,
                              size_t ws_size, hipStream_t stream) {
  (void)in_sizes; (void)n_in; (void)out_size; (void)ws_size;
  const float* x     = (const float*)d_in[0];
  const float* w     = (const float*)d_in[1];
  const float* bias  = (const float*)d_in[2];
  const float* chain = (const float*)d_in[3];
  const float* lb    = (const float*)d_in[4];
  const float* rb    = (const float*)d_in[5];
  float* pot = (float*)d_ws;          // 32768*64*4 = 8 MB scratch
  int* out = (int*)d_out;             // [B, T] int32 paths

  crf_potentials_gemm<<<dim3(MM / 128), dim3(256), 0, stream>>>(
      x, w, bias, lb, rb, pot);
  crf_viterbi<<<dim3(BB), dim3(256), 0, stream>>>(pot, chain, out);
}